// LinearChainCRF_20091857011539
// MI455X (gfx1250) — compile-verified
//
#include <hip/hip_runtime.h>
#include <math.h>

typedef float v2f __attribute__((ext_vector_type(2)));
typedef float v8f __attribute__((ext_vector_type(8)));

static constexpr int Bn  = 16;   // batch
static constexpr int Tn  = 512;  // time steps
static constexpr int Kn  = 128;  // states
static constexpr int LDP = 132;  // padded LDS row stride (bank-conflict free gathers)

__global__ __launch_bounds__(256, 1)
void crf_forward_kernel(const float* __restrict__ transition,
                        const float* __restrict__ emission,
                        const int*   __restrict__ seq_lens,
                        float* __restrict__ alpha_out,   // Bn*Tn*Kn
                        float* __restrict__ logZ_out)    // Bn
{
    __shared__ float sh_alpha[Bn * LDP];
    __shared__ float sh_u[Bn * LDP];
    __shared__ float sh_last[Bn * Kn];
    __shared__ float sh_amax[Bn];
    __shared__ float sh_part[Bn * 16];
    __shared__ int   sh_L[Bn];

    const int tid   = threadIdx.x;
    const int wave  = tid >> 5;          // 8 waves, wave w owns columns [16w, 16w+16)
    const int lane  = tid & 31;
    const int col   = (wave << 4) + (lane & 15); // j column of this lane in C tile
    const int khalf = lane >> 4;          // 0: lanes 0-15, 1: lanes 16-31
    const int arow  = lane & 15;          // A-matrix row (batch index) for this lane

    if (tid < Bn) sh_L[tid] = seq_lens[tid];

    // ---- Preload constant B fragments of E = exp(transition).
    // WMMA f32 16x16x4 B layout (4xK rows x 16 cols): VGPR0 -> rows {0,2} split across
    // lane halves, VGPR1 -> rows {1,3} (mirrors documented A/C striping).
    v2f Bfrag[32];
#pragma unroll
    for (int kk = 0; kk < 32; ++kk) {
        int r0 = (kk << 2) + (khalf << 1);
        Bfrag[kk].x = __expf(transition[r0 * Kn + col]);
        Bfrag[kk].y = __expf(transition[(r0 + 1) * Kn + col]);
    }

    // ---- t = 0: alpha0 = emission[:, 0, :]
#pragma unroll
    for (int r = 0; r < 8; ++r) {
        int b = r + (khalf << 3);
        float v = emission[((size_t)b * Tn + 0) * Kn + col];
        sh_alpha[b * LDP + col] = v;
        alpha_out[((size_t)b * Tn + 0) * Kn + col] = v;
    }
    __syncthreads();

    // capture rows with seq_len == 1
    {
        int e = tid << 3;
#pragma unroll
        for (int q = 0; q < 8; ++q) {
            int b = (e + q) >> 7, j = (e + q) & 127;
            if (sh_L[b] == 1) sh_last[b * Kn + j] = sh_alpha[b * LDP + j];
        }
    }
    __syncthreads();

    for (int t = 1; t < Tn; ++t) {
        // ---- amax[b] = max_j alpha[b,j] : 16 partials per row, then reduce
        {
            int row = tid >> 4, part = tid & 15;
            float m = -3.4e38f;
            int base = row * LDP + part * 8;
#pragma unroll
            for (int q = 0; q < 8; ++q) m = fmaxf(m, sh_alpha[base + q]);
            sh_part[row * 16 + part] = m;
        }
        __syncthreads();
        if (tid < Bn) {
            float m = -3.4e38f;
#pragma unroll
            for (int q = 0; q < 16; ++q) m = fmaxf(m, sh_part[tid * 16 + q]);
            sh_amax[tid] = m;
        }
        __syncthreads();

        // ---- u[b,i] = exp(alpha[b,i] - amax[b]) computed once (shared by all waves)
        {
            int e = tid << 3;
#pragma unroll
            for (int q = 0; q < 8; ++q) {
                int b = (e + q) >> 7, j = (e + q) & 127;
                sh_u[b * LDP + j] = __expf(sh_alpha[b * LDP + j] - sh_amax[b]);
            }
        }
        __syncthreads();

        // ---- V(16x128) = U(16x128) @ E(128x128): chain of 32 f32 WMMAs per wave
        v8f acc = {};
#pragma unroll
        for (int kk = 0; kk < 32; ++kk) {
            int i0 = (kk << 2) + (khalf << 1);   // A layout: lanes0-15 K={4kk,4kk+1}, lanes16-31 K={4kk+2,4kk+3}
            v2f a;
            a.x = sh_u[arow * LDP + i0];
            a.y = sh_u[arow * LDP + i0 + 1];
            acc = __builtin_amdgcn_wmma_f32_16x16x4_f32(
                false, a, false, Bfrag[kk], (short)0, acc, false, false);
        }

        // ---- alpha_t = emission + amax + log(V); write LDS + global
#pragma unroll
        for (int r = 0; r < 8; ++r) {
            int b = r + (khalf << 3);            // C/D layout: VGPR r -> M=r (+8 for upper lanes)
            float em = emission[((size_t)b * Tn + t) * Kn + col];
            float av = em + sh_amax[b] + __logf(acc[r]);
            sh_alpha[b * LDP + col] = av;
            alpha_out[((size_t)b * Tn + t) * Kn + col] = av;
        }
        __syncthreads();

        // ---- capture rows finishing at this t
        {
            int e = tid << 3;
#pragma unroll
            for (int q = 0; q < 8; ++q) {
                int b = (e + q) >> 7, j = (e + q) & 127;
                if (sh_L[b] - 1 == t) sh_last[b * Kn + j] = sh_alpha[b * LDP + j];
            }
        }
        __syncthreads();
    }

    // ---- log_Z[b] = logsumexp(last[b,:])
    if (tid < Bn) {
        float m = -3.4e38f;
        for (int j = 0; j < Kn; ++j) m = fmaxf(m, sh_last[tid * Kn + j]);
        float s = 0.f;
        for (int j = 0; j < Kn; ++j) s += __expf(sh_last[tid * Kn + j] - m);
        logZ_out[tid] = m + __logf(s);
    }
}

extern "C" void kernel_launch(void* const* d_in, const int* in_sizes, int n_in,
                              void* d_out, int out_size, void* d_ws, size_t ws_size,
                              hipStream_t stream) {
    const float* transition = (const float*)d_in[0];
    const float* emission   = (const float*)d_in[1];
    const int*   seq_lens   = (const int*)d_in[2];
    float* alpha_out = (float*)d_out;
    float* logZ_out  = (float*)d_out + (size_t)Bn * Tn * Kn;
    crf_forward_kernel<<<1, 256, 0, stream>>>(transition, emission, seq_lens,
                                              alpha_out, logZ_out);
}